// ICP_83107617177752
// MI455X (gfx1250) — compile-verified
//
#include <hip/hip_runtime.h>

// ICP (8 batches x 2048 pts, 8 iterations) in one persistent kernel:
// - one workgroup (1024 thr / 32 waves) per batch; everything LDS-resident
// - NN search via V_WMMA_F32_16X16X4_F32: A row = (-2x,-2y,-2z, 1),
//   B col = (tx,ty,tz,||t||^2)  =>  C[m][n] = ||t_n||^2 - 2 r_m . t_n
//   B fragments pre-packed in WMMA lane layout (target is invariant), so the
//   inner loop is ds_load_b64 + wmma + 8 cmp/select, branch-free.
// - per-lane running argmin over 128 column tiles, 16-lane shfl_xor reduce
// - 3x3 Kabsch via polar decomposition R = M (M^T M)^{-1/2} (Jacobi eigensolve)

typedef __attribute__((ext_vector_type(2))) float v2f;
typedef __attribute__((ext_vector_type(8))) float v8f;

#define NPTS   2048
#define NTILES (NPTS / 16)
#define BLOCK  1024
#define NWAVES (BLOCK / 32)
#define NITERS 8

__device__ inline void kabsch_rotation(const float Mm[3][3], float R[3][3]) {
    // A = Mm^T * Mm (symmetric PSD)
    float A[3][3];
#pragma unroll
    for (int a = 0; a < 3; ++a)
#pragma unroll
        for (int b = 0; b < 3; ++b)
            A[a][b] = Mm[0][a] * Mm[0][b] + Mm[1][a] * Mm[1][b] + Mm[2][a] * Mm[2][b];

    float V[3][3] = {{1.f, 0.f, 0.f}, {0.f, 1.f, 0.f}, {0.f, 0.f, 1.f}};
    // cyclic Jacobi sweeps to diagonalize A, accumulating V
#pragma unroll
    for (int sweep = 0; sweep < 8; ++sweep) {
#pragma unroll
        for (int pi = 0; pi < 3; ++pi) {
            const int p = (pi == 2) ? 1 : 0;
            const int q = (pi == 0) ? 1 : 2;
            float apq = A[p][q];
            float theta = 0.5f * atan2f(2.0f * apq, A[q][q] - A[p][p]);
            float c = __cosf(theta), s = __sinf(theta);
            float u0, u1;
#pragma unroll
            for (int k = 0; k < 3; ++k) {  // A <- A * J
                u0 = c * A[k][p] - s * A[k][q];
                u1 = s * A[k][p] + c * A[k][q];
                A[k][p] = u0; A[k][q] = u1;
            }
#pragma unroll
            for (int k = 0; k < 3; ++k) {  // A <- J^T * A
                u0 = c * A[p][k] - s * A[q][k];
                u1 = s * A[p][k] + c * A[q][k];
                A[p][k] = u0; A[q][k] = u1;
            }
#pragma unroll
            for (int k = 0; k < 3; ++k) {  // V <- V * J
                u0 = c * V[k][p] - s * V[k][q];
                u1 = s * V[k][p] + c * V[k][q];
                V[k][p] = u0; V[k][q] = u1;
            }
        }
    }
    // W = V * diag(1/sqrt(lam)) * V^T  == (M^T M)^{-1/2}
    float ws[3];
#pragma unroll
    for (int k = 0; k < 3; ++k) ws[k] = rsqrtf(fmaxf(A[k][k], 1e-12f));
    float W[3][3];
#pragma unroll
    for (int i = 0; i < 3; ++i)
#pragma unroll
        for (int j = 0; j < 3; ++j)
            W[i][j] = V[i][0] * ws[0] * V[j][0] + V[i][1] * ws[1] * V[j][1] +
                      V[i][2] * ws[2] * V[j][2];
    // R = M * W  (polar orthogonal factor == U V^T)
#pragma unroll
    for (int i = 0; i < 3; ++i)
#pragma unroll
        for (int j = 0; j < 3; ++j)
            R[i][j] = Mm[i][0] * W[0][j] + Mm[i][1] * W[1][j] + Mm[i][2] * W[2][j];
    // reference flips sign when det(R) ~ -1
    float det = R[0][0] * (R[1][1] * R[2][2] - R[1][2] * R[2][1]) -
                R[0][1] * (R[1][0] * R[2][2] - R[1][2] * R[2][0]) +
                R[0][2] * (R[1][0] * R[2][1] - R[1][1] * R[2][0]);
    if (fabsf(det + 1.0f) < 1e-6f) {
#pragma unroll
        for (int i = 0; i < 3; ++i)
#pragma unroll
            for (int j = 0; j < 3; ++j) R[i][j] = -R[i][j];
    }
}

__global__ __launch_bounds__(BLOCK) void icp_wmma_kernel(
    const float* __restrict__ src, const float* __restrict__ tgt,
    float* __restrict__ out) {
    // B fragments in WMMA lane layout, per 16-wide column tile:
    //   s_bf[t*32 + l]      (l<16)  = (tx, ty)        of target point t*16+l
    //   s_bf[t*32 + 16 + l] (l<16)  = (tz, ||t||^2)   of target point t*16+l
    __shared__ v2f            s_bf[NTILES * 32];           // 32 KB
    __shared__ float          s_px[NPTS], s_py[NPTS], s_pz[NPTS];  // 24 KB
    __shared__ unsigned short s_knn[NPTS];                 // 4 KB
    __shared__ float          s_red[16];
    __shared__ float          s_rt[12];

    const int b    = blockIdx.x;
    const int tid  = threadIdx.x;
    const int lane = tid & 31;
    const int wave = tid >> 5;
    const int half = lane >> 4;   // 0: lanes 0-15 (K=0,1), 1: lanes 16-31 (K=2,3)
    const int ln   = lane & 15;

    const float* sb = src + (size_t)b * NPTS * 3;
    const float* tb = tgt + (size_t)b * NPTS * 3;

    __builtin_prefetch(sb + tid * 6, 0, 3);   // global_prefetch_b8 -> warm L2
    __builtin_prefetch(tb + tid * 6, 0, 3);

    for (int i = tid; i < NPTS; i += BLOCK) {
        float tx = tb[i * 3 + 0], ty = tb[i * 3 + 1], tz = tb[i * 3 + 2];
        float tn = tx * tx + ty * ty + tz * tz;
        const int base = (i >> 4) * 32 + (i & 15);
        s_bf[base]      = (v2f){tx, ty};
        s_bf[base + 16] = (v2f){tz, tn};
        s_px[i] = sb[i * 3 + 0];
        s_py[i] = sb[i * 3 + 1];
        s_pz[i] = sb[i * 3 + 2];
    }
    __syncthreads();

    for (int it = 0; it < NITERS; ++it) {
        // ---- 1-NN search: WMMA f32 16x16x4 over 16x16 score tiles ----
        for (int rt = wave; rt < NTILES; rt += NWAVES) {
            const int arow = rt * 16 + ln;
            const float px = s_px[arow], py = s_py[arow], pz = s_pz[arow];
            v2f a;
            a.x = half ? (-2.0f * pz) : (-2.0f * px);
            a.y = half ? 1.0f : (-2.0f * py);

            float bestd[8];
            int   bestc[8];
#pragma unroll
            for (int v = 0; v < 8; ++v) { bestd[v] = 3.4e38f; bestc[v] = 0; }

#pragma unroll 2
            for (int t = 0; t < NTILES; ++t) {
                const v2f bv = s_bf[t * 32 + lane];
                v8f c = {};
                // c[v] = ||t_n||^2 - 2 r_m . t_n  (argmin-equivalent dist^2)
                c = __builtin_amdgcn_wmma_f32_16x16x4_f32(
                    false, a, false, bv, (short)0, c, false, false);
#pragma unroll
                for (int v = 0; v < 8; ++v) {
                    const bool lt = c[v] < bestd[v];
                    bestd[v] = lt ? c[v] : bestd[v];
                    bestc[v] = lt ? t : bestc[v];
                }
            }
            // reduce across the 16 lanes (columns) in each half-wave
#pragma unroll
            for (int v = 0; v < 8; ++v) {
                int col = bestc[v] * 16 + ln;
#pragma unroll
                for (int off = 8; off >= 1; off >>= 1) {
                    float od = __shfl_xor(bestd[v], off, 32);
                    int   oc = __shfl_xor(col, off, 32);
                    if (od < bestd[v]) { bestd[v] = od; col = oc; }
                }
                if (ln == 0) s_knn[rt * 16 + half * 8 + v] = (unsigned short)col;
            }
        }
        __syncthreads();

        // ---- centroid / cross-covariance sums: s=temporal, t=matched target ----
        if (tid < 16) s_red[tid] = 0.0f;
        __syncthreads();
        {
            float acc[15];
#pragma unroll
            for (int k = 0; k < 15; ++k) acc[k] = 0.0f;
            for (int i = tid; i < NPTS; i += BLOCK) {
                float sx = s_px[i], sy = s_py[i], sz = s_pz[i];
                const int j = s_knn[i];
                const int jb = (j >> 4) * 32 + (j & 15);
                v2f xy = s_bf[jb];
                v2f zn = s_bf[jb + 16];
                float tx = xy.x, ty = xy.y, tz = zn.x;
                acc[0] += sx;  acc[1] += sy;  acc[2] += sz;
                acc[3] += tx;  acc[4] += ty;  acc[5] += tz;
                acc[6]  += tx * sx; acc[7]  += tx * sy; acc[8]  += tx * sz;
                acc[9]  += ty * sx; acc[10] += ty * sy; acc[11] += ty * sz;
                acc[12] += tz * sx; acc[13] += tz * sy; acc[14] += tz * sz;
            }
#pragma unroll
            for (int k = 0; k < 15; ++k) {
#pragma unroll
                for (int off = 16; off >= 1; off >>= 1)
                    acc[k] += __shfl_xor(acc[k], off, 32);
                if (lane == 0) atomicAdd(&s_red[k], acc[k]);
            }
        }
        __syncthreads();

        // ---- Kabsch on thread 0, broadcast R|t via LDS ----
        if (tid == 0) {
            const float invN = 1.0f / (float)NPTS;
            float csv[3] = {s_red[0] * invN, s_red[1] * invN, s_red[2] * invN};
            float ctv[3] = {s_red[3] * invN, s_red[4] * invN, s_red[5] * invN};
            float Mm[3][3];
#pragma unroll
            for (int a2 = 0; a2 < 3; ++a2)
#pragma unroll
                for (int b2 = 0; b2 < 3; ++b2)
                    Mm[a2][b2] = s_red[6 + a2 * 3 + b2] -
                                 (float)NPTS * ctv[a2] * csv[b2];
            float R[3][3];
            kabsch_rotation(Mm, R);
#pragma unroll
            for (int i2 = 0; i2 < 3; ++i2) {
                s_rt[i2 * 3 + 0] = R[i2][0];
                s_rt[i2 * 3 + 1] = R[i2][1];
                s_rt[i2 * 3 + 2] = R[i2][2];
                s_rt[9 + i2] = ctv[i2] - (R[i2][0] * csv[0] + R[i2][1] * csv[1] +
                                          R[i2][2] * csv[2]);
            }
        }
        __syncthreads();

        // ---- apply p' = R p + t to temporal in LDS ----
        {
            float r00 = s_rt[0], r01 = s_rt[1], r02 = s_rt[2];
            float r10 = s_rt[3], r11 = s_rt[4], r12 = s_rt[5];
            float r20 = s_rt[6], r21 = s_rt[7], r22 = s_rt[8];
            float t0 = s_rt[9], t1 = s_rt[10], t2 = s_rt[11];
            for (int i = tid; i < NPTS; i += BLOCK) {
                float x = s_px[i], y = s_py[i], z = s_pz[i];
                s_px[i] = r00 * x + r01 * y + r02 * z + t0;
                s_py[i] = r10 * x + r11 * y + r12 * z + t1;
                s_pz[i] = r20 * x + r21 * y + r22 * z + t2;
            }
        }
        __syncthreads();
    }

    // ---- final Kabsch: source (global) -> temporal (LDS) ----
    if (tid < 16) s_red[tid] = 0.0f;
    __syncthreads();
    {
        float acc[15];
#pragma unroll
        for (int k = 0; k < 15; ++k) acc[k] = 0.0f;
        for (int i = tid; i < NPTS; i += BLOCK) {
            float sx = sb[i * 3 + 0], sy = sb[i * 3 + 1], sz = sb[i * 3 + 2];
            float tx = s_px[i], ty = s_py[i], tz = s_pz[i];
            acc[0] += sx;  acc[1] += sy;  acc[2] += sz;
            acc[3] += tx;  acc[4] += ty;  acc[5] += tz;
            acc[6]  += tx * sx; acc[7]  += tx * sy; acc[8]  += tx * sz;
            acc[9]  += ty * sx; acc[10] += ty * sy; acc[11] += ty * sz;
            acc[12] += tz * sx; acc[13] += tz * sy; acc[14] += tz * sz;
        }
#pragma unroll
        for (int k = 0; k < 15; ++k) {
#pragma unroll
            for (int off = 16; off >= 1; off >>= 1)
                acc[k] += __shfl_xor(acc[k], off, 32);
            if (lane == 0) atomicAdd(&s_red[k], acc[k]);
        }
    }
    __syncthreads();

    if (tid == 0) {
        const float invN = 1.0f / (float)NPTS;
        float csv[3] = {s_red[0] * invN, s_red[1] * invN, s_red[2] * invN};
        float ctv[3] = {s_red[3] * invN, s_red[4] * invN, s_red[5] * invN};
        float Mm[3][3];
#pragma unroll
        for (int a2 = 0; a2 < 3; ++a2)
#pragma unroll
            for (int b2 = 0; b2 < 3; ++b2)
                Mm[a2][b2] =
                    s_red[6 + a2 * 3 + b2] - (float)NPTS * ctv[a2] * csv[b2];
        float R[3][3];
        kabsch_rotation(Mm, R);
        float* ob = out + (size_t)b * 12;
#pragma unroll
        for (int i2 = 0; i2 < 3; ++i2) {
            ob[i2 * 4 + 0] = R[i2][0];
            ob[i2 * 4 + 1] = R[i2][1];
            ob[i2 * 4 + 2] = R[i2][2];
            ob[i2 * 4 + 3] = ctv[i2] - (R[i2][0] * csv[0] + R[i2][1] * csv[1] +
                                        R[i2][2] * csv[2]);
        }
    }
}

extern "C" void kernel_launch(void* const* d_in, const int* in_sizes, int n_in,
                              void* d_out, int out_size, void* d_ws,
                              size_t ws_size, hipStream_t stream) {
    const float* src = (const float*)d_in[0];
    const float* tgt = (const float*)d_in[1];
    float* out = (float*)d_out;
    const int B = in_sizes[0] / (NPTS * 3);
    icp_wmma_kernel<<<dim3(B), dim3(BLOCK), 0, stream>>>(src, tgt, out);
}